// QuantumConv_17772574671149
// MI455X (gfx1250) — compile-verified
//
#include <hip/hip_runtime.h>

// ---- problem constants (from reference) ----
#define NQ    4
#define CIN   64
#define HW    64
#define OUTC  128
#define DIN   576          // CIN * 3 * 3
#define TILE  16           // pixel tile is 16x16
#define CCH   16           // channel chunk loaded to LDS at a time
#define HALO  18           // TILE + 2 (3x3, pad 1)
#define CHUNK_ELEMS (CCH * HALO * HALO)

typedef float v2f __attribute__((ext_vector_type(2)));
typedef float v8f __attribute__((ext_vector_type(8)));

// ---- CDNA5 async global->LDS probe ----
#if __has_builtin(__builtin_amdgcn_global_load_async_to_lds_b32)
#define HAVE_ASYNC_LDS 1
typedef __attribute__((address_space(1))) int* g32p;   // printed as '__device__ int *'
typedef __attribute__((address_space(3))) int* l32p;   // LDS int*
#else
#define HAVE_ASYNC_LDS 0
#endif

__device__ __forceinline__ void wait_async_lds() {
#if HAVE_ASYNC_LDS
#if __has_builtin(__builtin_amdgcn_s_wait_asynccnt)
  __builtin_amdgcn_s_wait_asynccnt(0);
#else
  asm volatile("s_wait_asynccnt 0x0" ::: "memory");
#endif
#endif
}

// Stage one 16-channel chunk (with 3x3 halo, zero padded at image border) into LDS.
__device__ __forceinline__ void load_chunk(const float* __restrict__ x, float* dst,
                                           int b, int cc, int h0, int w0, int t) {
  for (int i = t; i < CHUNK_ELEMS; i += 256) {
    int c  = i / (HALO * HALO);
    int r  = i % (HALO * HALO);
    int ii = r / HALO, jj = r % HALO;
    int gh = h0 - 1 + ii, gw = w0 - 1 + jj;
    bool ok = (gh >= 0) & (gh < HW) & (gw >= 0) & (gw < HW);
    size_t gidx = (((size_t)b * CIN + (cc + c)) * HW + gh) * HW + gw;
#if HAVE_ASYNC_LDS
    if (ok) {
      __builtin_amdgcn_global_load_async_to_lds_b32(
          (g32p)(size_t)(const void*)&x[gidx],   // global (AS1) int*
          (l32p)(int*)&dst[i],                   // LDS (AS3) int*
          0, 0);
    } else {
      dst[i] = 0.0f;
    }
#else
    float v = 0.0f;
    if (ok) v = x[gidx];
    dst[i] = v;
#endif
  }
}

__global__ __launch_bounds__(256)
void qconv_fused(const float* __restrict__ x,
                 const float* __restrict__ fc1_w,
                 const float* __restrict__ fc1_b,
                 const float* __restrict__ u3p,
                 const float* __restrict__ cu3p,
                 const float* __restrict__ fc_w,
                 const float* __restrict__ fc_b,
                 float* __restrict__ out)
{
  __shared__ float  s_x[2][CHUNK_ELEMS];      // double-buffered x chunk (2 x 20.7 KB)
  __shared__ float4 s_w[DIN];                 // fc1_w transposed: s_w[k] = {w0k,w1k,w2k,w3k}
  __shared__ float  s_ez[TILE * TILE][NQ];    // expz staging for WMMA

  const int t   = threadIdx.x;
  const int b   = blockIdx.z;
  const int h0  = blockIdx.y * TILE;
  const int w0  = blockIdx.x * TILE;
  const int phh = t >> 4;          // pixel row in tile
  const int pww = t & 15;          // pixel col in tile

  // ---- preload fc1_w (transposed to float4-per-k) into LDS ----
  for (int k = t; k < DIN; k += 256)
    s_w[k] = make_float4(fc1_w[k], fc1_w[DIN + k], fc1_w[2 * DIN + k], fc1_w[3 * DIN + k]);

  float ang0 = fc1_b[0], ang1 = fc1_b[1], ang2 = fc1_b[2], ang3 = fc1_b[3];

  // ---- GEMM1 (3x3x64 conv -> 4 angles), double-buffered channel chunks ----
  load_chunk(x, s_x[0], b, 0, h0, w0, t);
  wait_async_lds();
  __syncthreads();

  for (int cc = 0; cc < CIN; cc += CCH) {
    const int cur = (cc / CCH) & 1;
    if (cc + CCH < CIN)                       // prefetch next chunk (async, overlaps compute)
      load_chunk(x, s_x[cur ^ 1], b, cc + CCH, h0, w0, t);

    const float* sx = s_x[cur];
    for (int c = 0; c < CCH; ++c) {
      #pragma unroll
      for (int di = 0; di < 3; ++di) {
        #pragma unroll
        for (int dj = 0; dj < 3; ++dj) {
          float  v  = sx[(c * HALO + (phh + di)) * HALO + (pww + dj)];
          float4 wv = s_w[(cc + c) * 9 + di * 3 + dj];   // uniform -> ds broadcast
          ang0 = fmaf(v, wv.x, ang0);
          ang1 = fmaf(v, wv.y, ang1);
          ang2 = fmaf(v, wv.z, ang2);
          ang3 = fmaf(v, wv.w, ang3);
        }
      }
    }

    wait_async_lds();   // next chunk fully in LDS
    __syncthreads();    // everyone done reading cur, next buffer ready
  }

  // ---- 4-qubit circuit in registers (wire w <-> index bit (3-w), i.e. mask 8>>w) ----
  struct c32 { float x, y; };
  auto cmul = [](c32 a, c32 b) -> c32 {
    return { a.x * b.x - a.y * b.y, a.x * b.y + a.y * b.x };
  };
  auto cadd = [](c32 a, c32 b) -> c32 { return { a.x + b.x, a.y + b.y }; };

  float aw[NQ] = { ang0, ang1, ang2, ang3 };
  float cw[NQ], sw[NQ];
  #pragma unroll
  for (int w = 0; w < NQ; ++w) { cw[w] = __cosf(aw[w] * 0.5f); sw[w] = __sinf(aw[w] * 0.5f); }

  c32 st[16];
  #pragma unroll
  for (int i = 0; i < 16; ++i) {
    float p = ((i >> 3) & 1 ? sw[0] : cw[0]) * ((i >> 2) & 1 ? sw[1] : cw[1]) *
              ((i >> 1) & 1 ? sw[2] : cw[2]) * ((i     ) & 1 ? sw[3] : cw[3]);
    st[i] = { p, 0.0f };
  }

  // trainable U3 on each wire
  #pragma unroll
  for (int w = 0; w < NQ; ++w) {
    float th = u3p[w * 3 + 0], ph = u3p[w * 3 + 1], la = u3p[w * 3 + 2];
    float ct = __cosf(th * 0.5f), s2 = __sinf(th * 0.5f);
    c32 U00 = { ct, 0.0f };
    c32 U01 = { -__cosf(la) * s2,      -__sinf(la) * s2 };
    c32 U10 = {  __cosf(ph) * s2,       __sinf(ph) * s2 };
    c32 U11 = {  __cosf(ph + la) * ct,  __sinf(ph + la) * ct };
    int m = 8 >> w;
    #pragma unroll
    for (int i = 0; i < 16; ++i) {
      if (i & m) continue;
      c32 a = st[i], bb = st[i | m];
      st[i]     = cadd(cmul(U00, a), cmul(U01, bb));
      st[i | m] = cadd(cmul(U10, a), cmul(U11, bb));
    }
  }

  // circular CU3 ring (w -> w+1 mod 4), applied on control=1 subspace
  #pragma unroll
  for (int w = 0; w < NQ; ++w) {
    float th = cu3p[w * 3 + 0], ph = cu3p[w * 3 + 1], la = cu3p[w * 3 + 2];
    float ct = __cosf(th * 0.5f), s2 = __sinf(th * 0.5f);
    c32 U00 = { ct, 0.0f };
    c32 U01 = { -__cosf(la) * s2,      -__sinf(la) * s2 };
    c32 U10 = {  __cosf(ph) * s2,       __sinf(ph) * s2 };
    c32 U11 = {  __cosf(ph + la) * ct,  __sinf(ph + la) * ct };
    int cm = 8 >> w, tm = 8 >> ((w + 1) & 3);
    #pragma unroll
    for (int i = 0; i < 16; ++i) {
      if (!(i & cm) || (i & tm)) continue;
      c32 a = st[i], bb = st[i | tm];
      st[i]      = cadd(cmul(U00, a), cmul(U01, bb));
      st[i | tm] = cadd(cmul(U10, a), cmul(U11, bb));
    }
  }

  // <Z_w>
  float ez[NQ] = { 0.f, 0.f, 0.f, 0.f };
  #pragma unroll
  for (int i = 0; i < 16; ++i) {
    float p = st[i].x * st[i].x + st[i].y * st[i].y;
    #pragma unroll
    for (int w = 0; w < NQ; ++w) ez[w] += (i & (8 >> w)) ? -p : p;
  }

  s_ez[t][0] = ez[0]; s_ez[t][1] = ez[1]; s_ez[t][2] = ez[2]; s_ez[t][3] = ez[3];
  __syncthreads();

  // ---- GEMM2: out[b, p, :] = expz @ fc_w^T + fc_b via V_WMMA_F32_16X16X4_F32 (K=4 exact) ----
  const int lane = t & 31;
  const int wv   = t >> 5;      // 8 waves -> 2 tile rows each
  const int lo   = lane & 15;
  const int hi   = lane >> 4;   // 0: K=0,1 half; 1: K=2,3 half

  // Hoist B operands (4x16 blocks of fc_w^T) and bias; invariant across both M-blocks.
  v2f   Bv[8];
  float cb[8];
  #pragma unroll
  for (int nb = 0; nb < 8; ++nb) {
    const int n = nb * 16 + lo;
    const float* wp = &fc_w[n * 4 + (hi ? 2 : 0)];
    Bv[nb][0] = wp[0]; Bv[nb][1] = wp[1];
    cb[nb] = fc_b[n];
  }

  #pragma unroll
  for (int rr = 0; rr < 2; ++rr) {
    const int prow = wv * 2 + rr;              // tile row (M-block of 16 pixels)
    v2f A;                                     // 16x4 f32 A operand
    const float* ep = &s_ez[prow * 16 + lo][hi ? 2 : 0];
    A[0] = ep[0]; A[1] = ep[1];

    const int gh = h0 + prow;
    #pragma unroll
    for (int nb = 0; nb < 8; ++nb) {
      v8f Cm;
      #pragma unroll
      for (int r = 0; r < 8; ++r) Cm[r] = cb[nb];

      v8f D = __builtin_amdgcn_wmma_f32_16x16x4_f32(
                  false, A, false, Bv[nb], (short)0, Cm, false, false);

      // D[r]: pixel m = r + 8*hi of this row, channel n = nb*16+lo.
      // Output is the raw view (B, P=4096, 128): idx = ((b*4096 + p)*128 + n)
      const size_t pbase = (size_t)b * 4096 + (size_t)gh * 64 + (w0 + 8 * hi);
      const size_t obase = pbase * 128 + (nb * 16 + lo);
      #pragma unroll
      for (int r = 0; r < 8; ++r)
        out[obase + (size_t)r * 128] = D[r];
    }
  }
}

extern "C" void kernel_launch(void* const* d_in, const int* in_sizes, int n_in,
                              void* d_out, int out_size, void* d_ws, size_t ws_size,
                              hipStream_t stream) {
  const float* x     = (const float*)d_in[0];
  const float* fc1_w = (const float*)d_in[1];
  const float* fc1_b = (const float*)d_in[2];
  const float* u3p   = (const float*)d_in[3];
  const float* cu3p  = (const float*)d_in[4];
  const float* fc_w  = (const float*)d_in[5];
  const float* fc_b  = (const float*)d_in[6];
  float* out = (float*)d_out;

  dim3 grid(HW / TILE, HW / TILE, 16);   // 4 x 4 x B
  dim3 block(256);                       // 8 wave32
  qconv_fused<<<grid, block, 0, stream>>>(x, fc1_w, fc1_b, u3p, cu3p, fc_w, fc_b, out);
}